// GCN_33457795236328
// MI455X (gfx1250) — compile-verified
//
#include <hip/hip_runtime.h>
#include <hip/hip_bf16.h>

typedef __attribute__((ext_vector_type(2))) float v2f;
typedef __attribute__((ext_vector_type(8))) float v8f;

#define N_USER 30000
#define N_ITEM 50000
#define N_TOT  80000
#define FEAT   512
#define HID    256
#define DLAT   64

__device__ __forceinline__ void atomic_add_f32(float* p, float v) {
    __hip_atomic_fetch_add(p, v, __ATOMIC_RELAXED, __HIP_MEMORY_SCOPE_AGENT);
}

// ---------------------------------------------------------------------------
// Fused MLP: x_items = leaky_relu(feat @ w0^T + b0) @ w1^T + b1
// One wave per 16-row tile, 4 waves / block, 16KB LDS per wave for the
// 16x256 hidden tile (C-layout -> A-layout relayout between the two GEMMs).
// ---------------------------------------------------------------------------
__global__ void mlp_kernel(const float* __restrict__ feat,
                           const float* __restrict__ w0, const float* __restrict__ b0,
                           const float* __restrict__ w1, const float* __restrict__ b1,
                           float* __restrict__ x) {
    extern __shared__ float lds[];               // 4 * 16 * 256 floats
    const int wave = threadIdx.x >> 5;
    const int lane = threadIdx.x & 31;
    const int tile = blockIdx.x * 4 + wave;      // uniform within the wave
    if (tile >= N_ITEM / 16) return;             // wave-uniform exit, EXEC stays all-1
    const int row0 = tile * 16;
    const int m  = lane & 15;                    // A-row / B-col index for this lane
    const int hi = lane >> 4;                    // K half select
    float* hid = lds + wave * 16 * HID;

    // ---- stage 1: hidden[16,256] = leaky_relu(feat_tile @ w0^T + b0) ----
    const float* arow = feat + (size_t)(row0 + m) * FEAT;
    for (int nt = 0; nt < HID / 16; ++nt) {
        v8f c = {};
        const float* brow = w0 + (size_t)(nt * 16 + m) * FEAT;  // B[k][n] = w0[n][k]
        for (int k = 0; k < FEAT; k += 4) {
            v2f a = *(const v2f*)(arow + k + 2 * hi);
            v2f b = *(const v2f*)(brow + k + 2 * hi);
            c = __builtin_amdgcn_wmma_f32_16x16x4_f32(false, a, false, b,
                                                      (short)0, c, false, false);
        }
#pragma unroll
        for (int v = 0; v < 8; ++v) {
            int mm = v + 8 * hi;
            int nn = nt * 16 + m;
            float t = c[v] + b0[nn];
            hid[mm * HID + nn] = (t > 0.0f) ? t : 0.01f * t;   // leaky_relu(0.01)
        }
    }

    // ---- stage 2: out[16,64] = hidden @ w1^T + b1 ----
    for (int nt = 0; nt < DLAT / 16; ++nt) {
        v8f c = {};
        const float* brow = w1 + (size_t)(nt * 16 + m) * HID;   // B[k][n] = w1[n][k]
        for (int k = 0; k < HID; k += 4) {
            v2f a = *(const v2f*)(hid + m * HID + k + 2 * hi);
            v2f b = *(const v2f*)(brow + k + 2 * hi);
            c = __builtin_amdgcn_wmma_f32_16x16x4_f32(false, a, false, b,
                                                      (short)0, c, false, false);
        }
#pragma unroll
        for (int v = 0; v < 8; ++v) {
            int mm = v + 8 * hi;
            int nn = nt * 16 + m;
            x[(size_t)(N_USER + row0 + mm) * DLAT + nn] = c[v] + b1[nn];
        }
    }
}

// ---------------------------------------------------------------------------
// GCNConv linear: h0 = x @ conv_w^T   (80000 x 64 @ 64 x 64), 8 waves/block
// ---------------------------------------------------------------------------
__global__ void lin_kernel(const float* __restrict__ xin,
                           const float* __restrict__ w,
                           float* __restrict__ hout) {
    const int wave = threadIdx.x >> 5;
    const int lane = threadIdx.x & 31;
    const int tile = blockIdx.x * 8 + wave;
    if (tile >= N_TOT / 16) return;              // wave-uniform
    const int row0 = tile * 16;
    const int m  = lane & 15;
    const int hi = lane >> 4;
    const float* arow = xin + (size_t)(row0 + m) * DLAT;
    for (int nt = 0; nt < DLAT / 16; ++nt) {
        v8f c = {};
        const float* brow = w + (size_t)(nt * 16 + m) * DLAT;   // B[k][n] = conv_w[n][k]
#pragma unroll
        for (int k = 0; k < DLAT; k += 4) {
            v2f a = *(const v2f*)(arow + k + 2 * hi);
            v2f b = *(const v2f*)(brow + k + 2 * hi);
            c = __builtin_amdgcn_wmma_f32_16x16x4_f32(false, a, false, b,
                                                      (short)0, c, false, false);
        }
#pragma unroll
        for (int v = 0; v < 8; ++v) {
            hout[(size_t)(row0 + v + 8 * hi) * DLAT + nt * 16 + m] = c[v];
        }
    }
}

// ---------------------------------------------------------------------------
// Small elementwise / graph kernels
// ---------------------------------------------------------------------------
__global__ void copy_pref_kernel(const float* __restrict__ pref, float* __restrict__ x, int n) {
    int i = blockIdx.x * blockDim.x + threadIdx.x;
    if (i < n) x[i] = pref[i];
}

__global__ void normalize_kernel(float* __restrict__ x) {
    int row  = blockIdx.x * (blockDim.x >> 5) + (threadIdx.x >> 5);
    int lane = threadIdx.x & 31;
    if (row >= N_TOT) return;
    float* p = x + (size_t)row * DLAT + lane * 2;
    float a = p[0], b = p[1];
    float s = a * a + b * b;
#pragma unroll
    for (int off = 16; off > 0; off >>= 1) s += __shfl_xor(s, off, 32);
    float inv = 1.0f / fmaxf(sqrtf(s), 1e-12f);
    p[0] = a * inv;
    p[1] = b * inv;
}

__global__ void deg_init_kernel(float* __restrict__ deg) {
    int i = blockIdx.x * blockDim.x + threadIdx.x;
    if (i < N_TOT) deg[i] = 1.0f;                // self-loop weight
}

__global__ void deg_acc_kernel(const int* __restrict__ ei, int E, float* __restrict__ deg) {
    int e = blockIdx.x * blockDim.x + threadIdx.x;
    if (e < E) atomic_add_f32(&deg[ei[E + e]], 1.0f);   // target = col
}

__global__ void dinv_kernel(const float* __restrict__ deg, float* __restrict__ dinv) {
    int i = blockIdx.x * blockDim.x + threadIdx.x;
    if (i < N_TOT) dinv[i] = rsqrtf(deg[i]);     // deg >= 1 always
}

// agg = self-loop contribution: h0[i]*dinv[i]^2
__global__ void agg_init_kernel(const float* __restrict__ h0, const float* __restrict__ dinv,
                                float* __restrict__ agg) {
    int idx = blockIdx.x * blockDim.x + threadIdx.x;
    if (idx >= N_TOT * DLAT) return;
    int i = idx >> 6;
    float d = dinv[i];
    agg[idx] = h0[idx] * d * d;
}

// one wave of 32 lanes per edge; each lane handles 2 of the 64 channels
__global__ void scatter_kernel(const int* __restrict__ ei, int E,
                               const float* __restrict__ h0,
                               const float* __restrict__ dinv,
                               float* __restrict__ agg) {
    long long gid = (long long)blockIdx.x * blockDim.x + threadIdx.x;
    int e    = (int)(gid >> 5);
    int lane = (int)(gid & 31);
    if (e >= E) return;
    int r = ei[e];
    int c = ei[E + e];
    float nrm = dinv[r] * dinv[c];
    const float* src = h0  + (size_t)r * DLAT + lane * 2;
    float*       dst = agg + (size_t)c * DLAT + lane * 2;
    atomic_add_f32(dst,     src[0] * nrm);
    atomic_add_f32(dst + 1, src[1] * nrm);
}

__global__ void bias_add_kernel(const float* __restrict__ agg, const float* __restrict__ b,
                                float* __restrict__ h) {
    int idx = blockIdx.x * blockDim.x + threadIdx.x;
    if (idx >= N_TOT * DLAT) return;
    h[idx] = agg[idx] + b[idx & (DLAT - 1)];
}

// out[0 : N_TOT*D] = x + h + h1 ; out[N_TOT*D : ] = preference
__global__ void final_kernel(const float* __restrict__ x, const float* __restrict__ h,
                             const float* __restrict__ h1, const float* __restrict__ pref,
                             float* __restrict__ out) {
    int idx = blockIdx.x * blockDim.x + threadIdx.x;
    const int xn = N_TOT * DLAT;
    const int tot = xn + N_USER * DLAT;
    if (idx >= tot) return;
    if (idx < xn) out[idx] = x[idx] + h[idx] + h1[idx];
    else          out[idx] = pref[idx - xn];
}

// ---------------------------------------------------------------------------
extern "C" void kernel_launch(void* const* d_in, const int* in_sizes, int n_in,
                              void* d_out, int out_size, void* d_ws, size_t ws_size,
                              hipStream_t stream) {
    const int*   ei    = (const int*)  d_in[0];   // edge_index [2,E]
    const float* feat  = (const float*)d_in[1];   // [N_ITEM, FEAT]
    const float* pref  = (const float*)d_in[2];   // [N_USER, D]
    const float* w0    = (const float*)d_in[3];   // [HID, FEAT]
    const float* b0    = (const float*)d_in[4];   // [HID]
    const float* w1    = (const float*)d_in[5];   // [D, HID]
    const float* b1    = (const float*)d_in[6];   // [D]
    const float* cw    = (const float*)d_in[7];   // [D, D]
    const float* cb    = (const float*)d_in[8];   // [D]
    float* out = (float*)d_out;

    const int E = in_sizes[0] / 2;
    const size_t XSZ = (size_t)N_TOT * DLAT;      // 5,120,000 floats

    float* ws   = (float*)d_ws;
    float* x    = ws;
    float* h0   = x    + XSZ;
    float* agg  = h0   + XSZ;
    float* h    = agg  + XSZ;
    float* h1   = h    + XSZ;
    float* deg  = h1   + XSZ;
    float* dinv = deg  + N_TOT;

    const int B = 256;
    const int nxd_blk = (int)((XSZ + B - 1) / B);

    // x = [preference ; MLP(features)]
    mlp_kernel<<<(N_ITEM / 16 + 3) / 4, 128, 4 * 16 * HID * sizeof(float), stream>>>(
        feat, w0, b0, w1, b1, x);
    copy_pref_kernel<<<(N_USER * DLAT + B - 1) / B, B, 0, stream>>>(pref, x, N_USER * DLAT);
    normalize_kernel<<<(N_TOT + 7) / 8, B, 0, stream>>>(x);

    // gcn_norm
    deg_init_kernel<<<(N_TOT + B - 1) / B, B, 0, stream>>>(deg);
    deg_acc_kernel<<<(E + B - 1) / B, B, 0, stream>>>(ei, E, deg);
    dinv_kernel<<<(N_TOT + B - 1) / B, B, 0, stream>>>(deg, dinv);

    const int sc_blk = (int)(((long long)E * 32 + B - 1) / B);

    // conv 1: h = scatter(norm * (x @ cw^T)) + cb
    lin_kernel<<<(N_TOT / 16 + 7) / 8, 256, 0, stream>>>(x, cw, h0);
    agg_init_kernel<<<nxd_blk, B, 0, stream>>>(h0, dinv, agg);
    scatter_kernel<<<sc_blk, B, 0, stream>>>(ei, E, h0, dinv, agg);
    bias_add_kernel<<<nxd_blk, B, 0, stream>>>(agg, cb, h);

    // conv 2: h1 = scatter(norm * (h @ cw^T)) + cb
    lin_kernel<<<(N_TOT / 16 + 7) / 8, 256, 0, stream>>>(h, cw, h0);
    agg_init_kernel<<<nxd_blk, B, 0, stream>>>(h0, dinv, agg);
    scatter_kernel<<<sc_blk, B, 0, stream>>>(ei, E, h0, dinv, agg);
    bias_add_kernel<<<nxd_blk, B, 0, stream>>>(agg, cb, h1);

    // x_hat = x + h + h1 ; concat preference
    const int tot = (int)XSZ + N_USER * DLAT;
    final_kernel<<<(tot + B - 1) / B, B, 0, stream>>>(x, h, h1, pref, out);
}